// AffineFlow_84327387889962
// MI455X (gfx1250) — compile-verified
//
#include <hip/hip_runtime.h>

typedef __attribute__((ext_vector_type(2))) float v2f;
typedef __attribute__((ext_vector_type(8))) float v8f;

#define DIM    127
#define DP1    128
#define COLS   16      // samples (columns) per workgroup
#define NTERMS 24      // Taylor terms; ||t*M|| <~ 1.2 -> 24! remainder << f32 eps

// Reciprocal table for Taylor coefficients, constant-folded at compile time.
__device__ __constant__ float INV_K[NTERMS + 1] = {
    0.0f,
    1.0f / 1.0f,  1.0f / 2.0f,  1.0f / 3.0f,  1.0f / 4.0f,
    1.0f / 5.0f,  1.0f / 6.0f,  1.0f / 7.0f,  1.0f / 8.0f,
    1.0f / 9.0f,  1.0f / 10.0f, 1.0f / 11.0f, 1.0f / 12.0f,
    1.0f / 13.0f, 1.0f / 14.0f, 1.0f / 15.0f, 1.0f / 16.0f,
    1.0f / 17.0f, 1.0f / 18.0f, 1.0f / 19.0f, 1.0f / 20.0f,
    1.0f / 21.0f, 1.0f / 22.0f, 1.0f / 23.0f, 1.0f / 24.0f
};

// Augmented generator M = [[A, b], [0, 0]], synthesized on the fly.
// row in [0,127], col in [0,127]; col==127 -> b column; row==127 -> zero row.
__device__ __forceinline__ float loadM(const float* __restrict__ A,
                                       const float* __restrict__ b,
                                       int row, int col) {
    if (row >= DIM) return 0.0f;
    if (col < DIM) return A[row * DIM + col];
    return b[row];
}

__global__ __launch_bounds__(256) void affine_flow_expmv(
    const float* __restrict__ t,   // (N,)
    const float* __restrict__ x,   // (N, 127)
    const float* __restrict__ A,   // (127, 127)
    const float* __restrict__ b,   // (127,)
    float* __restrict__ out,       // (N, 127)
    int N)
{
    // Double-buffered V block: 128 rows x 16 columns, f32. Row-major [row][col]
    // is bank-conflict free for both the B-frag reads and C-layout writes.
    __shared__ float Vbuf[2][DP1][COLS];

    const int lane = threadIdx.x & 31;
    const int wave = threadIdx.x >> 5;
    const int m0   = wave * 16;            // this wave's output row strip
    const int c0   = blockIdx.x * COLS;    // first sample of this block
    const int col  = lane & 15;            // column this lane owns (C/D layout)
    const int half = lane >> 4;            // 0: K/M group 0, 1: K/M group 1

    const int cIdx = (c0 + col < N) ? (c0 + col) : (N - 1);
    const float tcol = t[cIdx];

    // ---- Load this wave's 16x128 strip of M as 32 A-fragments (K=4 each) ----
    // f32 16x4 A layout: lanes 0-15 hold row m0+lane, VGPR0=K0,VGPR1=K1;
    //                    lanes 16-31 hold row m0+lane-16, VGPR0=K2,VGPR1=K3.
    v2f af[32];
    const int arow = m0 + col;
    #pragma unroll
    for (int kk = 0; kk < 32; ++kk) {
        const int acol = 4 * kk + 2 * half;
        af[kk].x = loadM(A, b, arow, acol);
        af[kk].y = loadM(A, b, arow, acol + 1);
    }

    // ---- Initialize V_0 = [x; 1] in LDS buffer 0 (cooperative fill) ----
    for (int idx = threadIdx.x; idx < DP1 * COLS; idx += 256) {
        const int r = idx >> 4;
        const int c = idx & 15;
        const int n = (c0 + c < N) ? (c0 + c) : (N - 1);
        Vbuf[0][r][c] = (r < DIM) ? x[(size_t)n * DIM + r] : 1.0f;
    }
    __syncthreads();

    // ---- Y accumulator starts at V_0 (this wave's C/D-layout positions) ----
    // C/D f32 16x16 layout: lane group half h, VGPR g -> row m0 + 8*h + g.
    v8f y;
    #pragma unroll
    for (int g = 0; g < 8; ++g)
        y[g] = Vbuf[0][m0 + 8 * half + g][col];

    // ---- Taylor loop: V_k = (t/k) * (M @ V_{k-1}); Y += V_k ----
    int p = 0;
    for (int k = 1; k <= NTERMS; ++k) {
        v8f c = {};
        #pragma unroll
        for (int kk = 0; kk < 32; ++kk) {
            // B-frag (4x16): lanes 0-15 VGPR0=K row 4kk+0, VGPR1=K row 4kk+1;
            //                lanes 16-31 rows 4kk+2, 4kk+3.
            const int r = 4 * kk + 2 * half;
            v2f bf;
            bf.x = Vbuf[p][r][col];
            bf.y = Vbuf[p][r + 1][col];
            c = __builtin_amdgcn_wmma_f32_16x16x4_f32(
                    /*neg_a=*/false, af[kk],
                    /*neg_b=*/false, bf,
                    /*c_mod=*/(short)0, c,
                    /*reuse_a=*/false, /*reuse_b=*/false);
        }

        const float s = tcol * INV_K[k];   // per-column Taylor coefficient
        #pragma unroll
        for (int g = 0; g < 8; ++g) {
            c[g] *= s;
            y[g] += c[g];
            Vbuf[1 - p][m0 + 8 * half + g][col] = c[g];  // publish V_k
        }
        p ^= 1;
        __syncthreads();   // writes of V_k visible before next term's reads
    }

    // ---- Store y (drop homogeneous row 127) ----
    if (c0 + col < N) {
        #pragma unroll
        for (int g = 0; g < 8; ++g) {
            const int row = m0 + 8 * half + g;
            if (row < DIM)
                out[(size_t)(c0 + col) * DIM + row] = y[g];
        }
    }
}

extern "C" void kernel_launch(void* const* d_in, const int* in_sizes, int n_in,
                              void* d_out, int out_size, void* d_ws, size_t ws_size,
                              hipStream_t stream) {
    const float* t = (const float*)d_in[0];
    const float* x = (const float*)d_in[1];
    const float* A = (const float*)d_in[2];
    const float* b = (const float*)d_in[3];
    float* out = (float*)d_out;

    const int N = in_sizes[0];                 // 4096
    const int nblocks = (N + COLS - 1) / COLS; // 256

    affine_flow_expmv<<<nblocks, 256, 0, stream>>>(t, x, A, b, out, N);
}